// DeltaNetLayer_54150947668392
// MI455X (gfx1250) — compile-verified
//
#include <hip/hip_runtime.h>
#include <math.h>

// Problem constants (match reference)
#define B_ 4
#define T_ 2048
#define D_ 256
#define H_ 64
#define BT_ (B_ * T_)
#define LN_EPS 1e-5f

typedef float v2f __attribute__((ext_vector_type(2)));
typedef float v8f __attribute__((ext_vector_type(8)));

// -------------------------------------------------------------------------
// CDNA5 async global->LDS copy (ASYNCcnt-tracked), CDNA5 ISA.
// GV mode: 64-bit global address in VGPR pair, LDS byte address in VDST VGPR.
// Generic pointers to LDS carry the LDS byte offset in their low 32 bits.
// -------------------------------------------------------------------------
__device__ __forceinline__ void async_load_f32(float* lds_dst, const float* gsrc) {
    unsigned lds = (unsigned)(unsigned long long)lds_dst;
    asm volatile("global_load_async_to_lds_b32 %0, %1, off"
                 :: "v"(lds), "v"(gsrc)
                 : "memory");
}

__device__ __forceinline__ void wait_async0() {
#if __has_builtin(__builtin_amdgcn_s_wait_asynccnt)
    __builtin_amdgcn_s_wait_asynccnt(0);
#else
    asm volatile("s_wait_asynccnt 0x0" ::: "memory");
#endif
}

// -------------------------------------------------------------------------
// Kernel 1: fused projections q/k/v = x W^T + b (WMMA f32 16x16x4),
//           h = relu(x Wb1^T + bb1), beta = sigmoid(h Wb2^T + bb2).
// One block per 16-row tile of x; 8 waves; each wave owns 16x16 output tiles.
// -------------------------------------------------------------------------
#define XS_STRIDE 260   // bank = (4*m + k) % 64, conflict-free A-fragment reads
#define HS_STRIDE 68

__global__ __launch_bounds__(256)
void proj_kernel(const float* __restrict__ x,
                 const float* __restrict__ Wq, const float* __restrict__ bq,
                 const float* __restrict__ Wk, const float* __restrict__ bk,
                 const float* __restrict__ Wv, const float* __restrict__ bv,
                 const float* __restrict__ Wb1, const float* __restrict__ bb1,
                 const float* __restrict__ Wb2, const float* __restrict__ bb2,
                 float* __restrict__ qo, float* __restrict__ ko,
                 float* __restrict__ vo, float* __restrict__ betao)
{
    __shared__ float xs[16 * XS_STRIDE];
    __shared__ float hs[16 * HS_STRIDE];

    const int t    = threadIdx.x;
    const int wave = t >> 5;
    const int lane = t & 31;
    const int row0 = blockIdx.x * 16;

    // Stage the 16x256 x-tile into LDS.
    {
        const int m  = t >> 4;
        const int c0 = (t & 15) * 16;
        const float* src = x + (size_t)(row0 + m) * D_ + c0;
        #pragma unroll
        for (int i = 0; i < 16; ++i)
            xs[m * XS_STRIDE + c0 + i] = src[i];
    }
    __syncthreads();

    // 52 output tiles: 0-15 -> q, 16-31 -> k, 32-47 -> v, 48-51 -> h.
    const int ml    = lane & 15;         // N (B/C/D) and M (A)
    const int kh    = (lane >> 4) * 2;   // K sub-offset for A/B fragments
    const int mbase = (lane >> 4) * 8;   // C/D row base

    for (int tile = wave; tile < 52; tile += 8) {
        const float* W; const float* bias; int n0;
        if (tile < 16)      { W = Wq;  bias = bq;  n0 = tile * 16;        }
        else if (tile < 32) { W = Wk;  bias = bk;  n0 = (tile - 16) * 16; }
        else if (tile < 48) { W = Wv;  bias = bv;  n0 = (tile - 32) * 16; }
        else                { W = Wb1; bias = bb1; n0 = (tile - 48) * 16; }

        // A[m][k] = xs[m][k]; B[k][n] = W[n][k]  (so D = x @ W^T)
        const float* arow = &xs[ml * XS_STRIDE];
        const float* wrow = W + (size_t)(n0 + ml) * D_;

        v8f acc = {};
        #pragma unroll 8
        for (int k0 = 0; k0 < D_; k0 += 4) {
            v2f a = *(const v2f*)(arow + k0 + kh);
            v2f b = *(const v2f*)(wrow + k0 + kh);
            acc = __builtin_amdgcn_wmma_f32_16x16x4_f32(
                false, a, false, b, (short)0, acc, false, false);
        }

        const float bval = bias[n0 + ml];
        if (tile < 48) {
            float* outp = (tile < 16) ? qo : (tile < 32 ? ko : vo);
            #pragma unroll
            for (int i = 0; i < 8; ++i)
                outp[(size_t)(row0 + mbase + i) * D_ + n0 + ml] = acc[i] + bval;
        } else {
            #pragma unroll
            for (int i = 0; i < 8; ++i) {
                float r = acc[i] + bval;
                hs[(mbase + i) * HS_STRIDE + n0 + ml] = r > 0.f ? r : 0.f;
            }
        }
    }
    __syncthreads();

    // beta = sigmoid(h @ Wb2^T + bb2): 16 rows, 64-length dot each.
    if (t < 16) {
        float acc = bb2[0];
        #pragma unroll 8
        for (int j = 0; j < H_; ++j)
            acc += hs[t * HS_STRIDE + j] * Wb2[j];
        betao[row0 + t] = 1.f / (1.f + __expf(-acc));
    }
}

// -------------------------------------------------------------------------
// Kernel 2: sequential DeltaNet scan, one block (32 wave32s) per batch.
// S (256 rows x 260-float padded stride = 266KB) lives in LDS (320KB/WGP).
// Thread t owns S[row=t>>2][cols quad + 4*j], so a wave's lanes hit banks
// (4*row + quad + 4*j) % 64: all 32 distinct -> conflict-free.
// Per-step q/k/v/beta are double-buffered in LDS and prefetched for step
// t+1 with async global->LDS loads overlapped with step t's compute.
// -------------------------------------------------------------------------
#define S_STRIDE  260
#define VEC_STRIDE 772   // k[0..255] q[256..511] v[512..767] beta[768] +pad

__global__ __launch_bounds__(1024, 1)
void scan_kernel(const float* __restrict__ qg, const float* __restrict__ kg,
                 const float* __restrict__ vg, const float* __restrict__ betag,
                 const float* __restrict__ gamma, const float* __restrict__ beta_ln,
                 float* __restrict__ out)
{
    extern __shared__ float smem[];
    float* S     = smem;                        // 256*260 = 66560 floats
    float* vbuf0 = smem + 256 * S_STRIDE;       // VEC_STRIDE
    float* vbuf1 = vbuf0 + VEC_STRIDE;          // VEC_STRIDE
    float* ov    = vbuf1 + VEC_STRIDE;          // 256
    float* gam   = ov + 256;                    // 256
    float* bln   = gam + 256;                   // 256
    float* stats = bln + 256;                   // [0]=mu [1]=rstd

    const int t    = threadIdx.x;
    const int b    = blockIdx.x;
    const int row  = t >> 2;
    const int quad = t & 3;
    float* Srow = S + row * S_STRIDE + quad;    // elements at Srow[4*j]

    const float* qb = qg + (size_t)b * T_ * D_;
    const float* kb = kg + (size_t)b * T_ * D_;
    const float* vb = vg + (size_t)b * T_ * D_;

    // Prefetch step 0 vectors (overlaps with the big LDS zero-fill below).
    if (t < 256)       async_load_f32(vbuf0 + t, kb + t);
    else if (t < 512)  async_load_f32(vbuf0 + t, qb + (t - 256));
    else if (t < 768)  async_load_f32(vbuf0 + t, vb + (t - 512));
    else if (t == 768) async_load_f32(vbuf0 + 768, betag + (size_t)b * T_);

    for (int i = t; i < 256 * S_STRIDE; i += 1024) S[i] = 0.f;
    if (t < 256) { gam[t] = gamma[t]; bln[t] = beta_ln[t]; }
    wait_async0();
    __syncthreads();

    for (int ts = 0; ts < T_; ++ts) {
        float* cur = (ts & 1) ? vbuf1 : vbuf0;
        float* nxt = (ts & 1) ? vbuf0 : vbuf1;

        // Kick off async prefetch of step ts+1 (hidden behind compute).
        if (ts + 1 < T_) {
            const size_t off = (size_t)(ts + 1) * D_;
            if (t < 256)       async_load_f32(nxt + t, kb + off + t);
            else if (t < 512)  async_load_f32(nxt + t, qb + off + (t - 256));
            else if (t < 768)  async_load_f32(nxt + t, vb + off + (t - 512));
            else if (t == 768) async_load_f32(nxt + 768, betag + (size_t)b * T_ + ts + 1);
        }

        const float* kv = cur;
        const float* qv = cur + 256;
        const float beta_t = cur[768];

        // Sk[row] partial dot over this thread's interleaved 64 columns
        float p = 0.f;
        #pragma unroll 16
        for (int j = 0; j < 64; ++j) p += Srow[4 * j] * kv[quad + 4 * j];
        p += __shfl_xor(p, 1, 32);
        p += __shfl_xor(p, 2, 32);
        const float delta = beta_t * (cur[512 + row] - p);

        // Rank-1 update fused with o = S_new @ q partial dot
        float po = 0.f;
        #pragma unroll 16
        for (int j = 0; j < 64; ++j) {
            float s = Srow[4 * j] + delta * kv[quad + 4 * j];
            Srow[4 * j] = s;
            po += s * qv[quad + 4 * j];
        }
        po += __shfl_xor(po, 1, 32);
        po += __shfl_xor(po, 2, 32);
        if (quad == 0) ov[row] = po;
        __syncthreads();

        // LayerNorm stats by wave 0
        if (t < 32) {
            float s1 = 0.f, s2 = 0.f;
            #pragma unroll
            for (int i = 0; i < 8; ++i) {
                float v = ov[t * 8 + i];
                s1 += v; s2 += v * v;
            }
            #pragma unroll
            for (int m = 16; m >= 1; m >>= 1) {
                s1 += __shfl_xor(s1, m, 32);
                s2 += __shfl_xor(s2, m, 32);
            }
            if (t == 0) {
                float mu  = s1 * (1.f / 256.f);
                float var = s2 * (1.f / 256.f) - mu * mu;
                stats[0] = mu;
                stats[1] = rsqrtf(var + LN_EPS);
            }
        }
        __syncthreads();

        if (t < 256) {
            float v = (ov[t] - stats[0]) * stats[1];
            out[((size_t)b * T_ + ts) * D_ + t] = gam[t] * v + bln[t];
        }

        // Drain this step's prefetch, then release buffers for next step.
        wait_async0();
        __syncthreads();
    }
}

// -------------------------------------------------------------------------
extern "C" void kernel_launch(void* const* d_in, const int* in_sizes, int n_in,
                              void* d_out, int out_size, void* d_ws, size_t ws_size,
                              hipStream_t stream) {
    const float* x       = (const float*)d_in[0];
    const float* Wq      = (const float*)d_in[1];
    const float* bq      = (const float*)d_in[2];
    const float* Wk      = (const float*)d_in[3];
    const float* bk      = (const float*)d_in[4];
    const float* Wv      = (const float*)d_in[5];
    const float* bv      = (const float*)d_in[6];
    const float* Wb1     = (const float*)d_in[7];
    const float* bb1     = (const float*)d_in[8];
    const float* Wb2     = (const float*)d_in[9];
    const float* bb2     = (const float*)d_in[10];
    const float* gamma   = (const float*)d_in[11];
    const float* beta_ln = (const float*)d_in[12];
    float* out = (float*)d_out;

    float* ws = (float*)d_ws;
    float* q    = ws;
    float* k    = ws + (size_t)BT_ * D_;
    float* v    = ws + 2 * (size_t)BT_ * D_;
    float* beta = ws + 3 * (size_t)BT_ * D_;

    proj_kernel<<<BT_ / 16, 256, 0, stream>>>(
        x, Wq, bq, Wk, bk, Wv, bv, Wb1, bb1, Wb2, bb2, q, k, v, beta);

    const size_t smem_bytes =
        (256 * S_STRIDE + 2 * VEC_STRIDE + 3 * 256 + 8) * sizeof(float); // ~276KB of 320KB LDS
    scan_kernel<<<B_, 1024, smem_bytes, stream>>>(
        q, k, v, beta, gamma, beta_ln, out);
}